// diffNet_26895085207978
// MI455X (gfx1250) — compile-verified
//
#include <hip/hip_runtime.h>
#include <stdint.h>

#define N_TAB       256
#define DEPTH_STEPS 128
#define DT_F        0.001f
#define TPB         256    // 8 wave32s per block

// Each thread integrates TWO consecutive particles (one float4 of (x0,v0,x1,v1)),
// giving two independent dependency chains per lane and b128 global accesses.
__global__ __launch_bounds__(TPB)
void diffnet_scan_kernel(const float4* __restrict__ X4,
                         const float* __restrict__ force,
                         float4* __restrict__ out4,
                         int Bpairs)
{
    __shared__ float sforce[N_TAB + 1];

    // ---- Stage the 257-entry force LUT into LDS via CDNA5 async global->LDS DMA.
    {
        const uint32_t t    = threadIdx.x;
        const uint32_t lds0 = (uint32_t)(uintptr_t)(&sforce[0]);
        uint32_t ldsa = lds0 + t * 4u;
        uint32_t goff = t * 4u;
        asm volatile("global_load_async_to_lds_b32 %0, %1, %2"
                     :: "v"(ldsa), "v"(goff), "s"(force) : "memory");
        if (t == 0) {
            uint32_t ldsl = lds0 + (uint32_t)N_TAB * 4u;
            uint32_t gofl = (uint32_t)N_TAB * 4u;
            asm volatile("global_load_async_to_lds_b32 %0, %1, %2"
                         :: "v"(ldsl), "v"(gofl), "s"(force) : "memory");
        }
        asm volatile("s_wait_asynccnt 0" ::: "memory");
    }
    __syncthreads();

    const int pid = blockIdx.x * TPB + threadIdx.x;
    if (pid >= Bpairs) return;

    float4 p = X4[pid];                 // global_load_b128: (x0, v0, x1, v1)
    float x[2] = { p.x, p.z };
    float v[2] = { p.y, p.w };

    // Per-chain cached LUT cell: base value f0, slope d = f1-f0, cell origin fi
    // (as float), and cell end ce = fi+1. Valid while v < ce (v is monotone
    // non-decreasing because force >= 0).
    float f0[2], d[2], fi[2], ce[2];

    #pragma unroll
    for (int k = 0; k < 2; ++k) {
        x[k] = fmaf(DT_F, v[k], x[k]);  // x = x + DT*v (pre-scan update)
        float fl = floorf(v[k]);
        int   i0 = (int)fl;
        float a  = sforce[i0];
        float b  = sforce[min(i0 + 1, N_TAB)];
        f0[k] = a;  d[k] = b - a;  fi[k] = fl;  ce[k] = fl + 1.0f;
    }

    // ---- 128-step scan. Steady state per chain: sub, fma, fma, cmp, fma.
    // |dv| <= DT per step, so a cell crossing is rare; both chains share one
    // EXEC-masked reload region (s_cbranch_execz skips it).
    #pragma unroll 8
    for (int s = 0; s < DEPTH_STEPS; ++s) {
        #pragma unroll
        for (int k = 0; k < 2; ++k) {
            float alpha = v[k] - fi[k];                 // v - floor(v) while in cell
            float f     = fmaf(alpha, d[k], f0[k]);     // lerp(force, v)
            v[k] = fmaf(DT_F, f, v[k]);
            x[k] = fmaf(DT_F, v[k], x[k]);
        }
        if ((v[0] >= ce[0]) || (v[1] >= ce[1])) {       // rare: refresh both chains
            #pragma unroll
            for (int k = 0; k < 2; ++k) {
                float fl = floorf(v[k]);
                int   i0 = (int)fl;
                float a  = sforce[i0];
                float b  = sforce[min(i0 + 1, N_TAB)];
                f0[k] = a;  d[k] = b - a;  fi[k] = fl;  ce[k] = fl + 1.0f;
            }
        }
    }

    out4[pid] = make_float4(x[0], v[0], x[1], v[1]);    // global_store_b128
}

extern "C" void kernel_launch(void* const* d_in, const int* in_sizes, int n_in,
                              void* d_out, int out_size, void* d_ws, size_t ws_size,
                              hipStream_t stream) {
    const float4* X4    = (const float4*)d_in[0];   // (B, 2) f32 -> B/2 float4s
    const float*  force = (const float*)d_in[1];    // 257 f32
    float4*       out4  = (float4*)d_out;

    const int B      = in_sizes[0] / 2;             // B = 1048576 (even)
    const int Bpairs = B / 2;
    const int blocks = (Bpairs + TPB - 1) / TPB;    // 2048 blocks of 8 wave32s

    hipLaunchKernelGGL(diffnet_scan_kernel, dim3(blocks), dim3(TPB), 0, stream,
                       X4, force, out4, Bpairs);
}